// YoloLoss_84911503442343
// MI455X (gfx1250) — compile-verified
//
#include <hip/hip_runtime.h>
#include <hip/hip_bf16.h>
#include <math.h>

// ---------------- problem constants ----------------
#define AN      3
#define HH      128
#define WW      128
#define CC      2
#define ATTRS   7
#define BB      64
#define NCELLS  (BB * AN * HH * WW)        // 3,145,728
#define TILE    1024                       // cells per tile (28 KB)
#define NTILES  (NCELLS / TILE)            // 3,072
#define NBLK    1536                       // blocks (persistent)
#define NITER   (NTILES / NBLK)            // 2 tiles per block
#define TILES_PER_B (AN * HH * WW / TILE)  // 48
#define TGT_OFF (NBLK * 6)                 // float offset of target records in ws
#define TREC    12                         // floats per target record

typedef __attribute__((ext_vector_type(2))) float v2f;
typedef __attribute__((ext_vector_type(8))) float v8f;

// ---------------- WMMA wave32 reduction ----------------
// Sum s across all 32 lanes using two chained V_WMMA_F32_16X16X4_F32.
// A-operand (documented 16x4 layout): lane m holds row m (K=0,1), lane m+16
// holds row m (K=2,3). B = all-ones (layout invariant). D[m,n] = rowsum(m).
// Stage 1: rowsum(m) = 2*(s_m + s_{m+16}). D layout (documented): vgpr v =
// rowsum(v) on lanes 0-15, rowsum(v+8) on lanes 16-31. Collapsing even/odd
// vgprs into a second A operand makes stage 2 produce 2*total in every lane.
#if __has_builtin(__builtin_amdgcn_wmma_f32_16x16x4_f32)
__device__ __forceinline__ float waveReduce(float s) {
  v2f a;   a[0] = s;    a[1] = s;
  v2f one; one[0] = 1.f; one[1] = 1.f;
  v8f c = {};
  v8f d = __builtin_amdgcn_wmma_f32_16x16x4_f32(false, a, false, one,
                                                (short)0, c, false, false);
  v2f a2; a2[0] = d[0] + d[2] + d[4] + d[6];
          a2[1] = d[1] + d[3] + d[5] + d[7];
  v8f d2 = __builtin_amdgcn_wmma_f32_16x16x4_f32(false, a2, false, one,
                                                 (short)0, c, false, false);
  return d2[0] * 0.5f;
}
#else
__device__ __forceinline__ float waveReduce(float s) {
  for (int off = 16; off > 0; off >>= 1) s += __shfl_xor(s, off, 32);
  return s;
}
#endif

// ---------------- kernel 1: per-target preprocessing ----------------
__global__ void yolo_setup(const float* __restrict__ boxes,
                           const int*   __restrict__ labels,
                           const float* __restrict__ areas,
                           float* __restrict__ ws) {
  int t = threadIdx.x;                 // t = b*2 + i
  if (t >= BB * 2) return;
  int i = t & 1;
  const float* bx = boxes + (size_t)t * 4;
  float x1 = bx[0], y1 = bx[1], x2 = bx[2], y2 = bx[3];
  float cx = (x1 + x2) * 0.5f, cy = (y1 + y2) * 0.5f;
  float bw = x2 - x1,          bh = y2 - y1;
  float ar = areas[t];
  // argmin |anchor_area - area|, first index wins ties
  float d0 = fabsf(10440.f - ar), d1 = fabsf(30888.f - ar), d2 = fabsf(121598.f - ar);
  int anc = 0; float bd = d0;
  if (d1 < bd) { anc = 1; bd = d1; }
  if (d2 < bd) { anc = 2; }
  int cellX = (int)((bw - cx) * 0.125f);   // STRIDE_W = 8, trunc toward zero
  int cellY = (int)((bh - cy) * 0.125f);
  if (cellX < 0 || cellX >= HH || cellY < 0 || cellY >= WW) anc = -1; // JAX drops OOB scatter
  int lb  = labels[t];
  int idx = (i == 0) ? lb : lb - 1;
  float cls5  = (idx == 1) ? 1.f : 0.f;    // one_hot(idx, C=2) component 1 (full[...,5])
  float areaT = (bw - cx) * (bh - cy);     // (tx2-tx1)*(ty2-ty1) as in reference
  float* r = ws + TGT_OFF + t * TREC;
  r[0] = cx; r[1] = cy; r[2] = bw; r[3] = bh; r[4] = areaT; r[5] = cls5;
  r[6] = __int_as_float(anc);
  r[7] = __int_as_float(cellX);
  r[8] = __int_as_float(cellY);
}

// ---------------- kernel 2: streaming pass + reduction ----------------
__global__ __launch_bounds__(256) void yolo_main(const float* __restrict__ out,
                                                 float* __restrict__ ws) {
  __shared__ float wsum[8 * 6];
  const int tid = threadIdx.x;
  float s0 = 0.f, s1 = 0.f, s2 = 0.f, s3 = 0.f, s4 = 0.f, s5 = 0.f;
  const float* tg = ws + TGT_OFF;

  for (int it = 0; it < NITER; ++it) {
    const int tile   = blockIdx.x + it * NBLK;
    const int b      = tile / TILES_PER_B;
    const int within = tile - b * TILES_PER_B;
    const int cib    = within * TILE + tid * 4;     // cell index within batch b
    const int a      = cib >> 14;                   // / (H*W)
    const int hw     = cib & 16383;
    const int h      = hw >> 7;
    const int w0     = hw & 127;                    // 4 cells: w0..w0+3 (4-aligned)

    const size_t cellBase = (size_t)tile * TILE + (size_t)tid * 4;
    const float4* pf = (const float4*)(out + cellBase * ATTRS); // 16B aligned
    float v[28];
    *(float4*)(v +  0) = pf[0];
    *(float4*)(v +  4) = pf[1];
    *(float4*)(v +  8) = pf[2];
    *(float4*)(v + 12) = pf[3];
    *(float4*)(v + 16) = pf[4];
    *(float4*)(v + 20) = pf[5];
    *(float4*)(v + 24) = pf[6];
    if (it + 1 < NITER)  // prefetch next grid-stride tile (global_prefetch_b8)
      __builtin_prefetch(out + (cellBase + (size_t)NBLK * TILE) * ATTRS, 0, 0);

    // two targets for this batch (uniform per block -> scalar cached)
    const float* r0 = tg + (size_t)(b * 2) * TREC;
    const float cx0 = r0[0], cy0 = r0[1], bw0 = r0[2], bh0 = r0[3], aT0 = r0[4], cl0 = r0[5];
    const int anc0 = __float_as_int(r0[6]), cX0 = __float_as_int(r0[7]), cY0 = __float_as_int(r0[8]);
    const float* r1 = r0 + TREC;
    const float cx1 = r1[0], cy1 = r1[1], bw1 = r1[2], bh1 = r1[3], aT1 = r1[4], cl1 = r1[5];
    const int anc1 = __float_as_int(r1[6]), cX1 = __float_as_int(r1[7]), cY1 = __float_as_int(r1[8]);

#pragma unroll
    for (int j = 0; j < 4; ++j) {
      const int w = w0 + j;
      const float o0 = v[j * 7 + 0], o1 = v[j * 7 + 1], o2 = v[j * 7 + 2];
      const float o3 = v[j * 7 + 3], o4 = v[j * 7 + 4], o5 = v[j * 7 + 5];
      const float o6 = v[j * 7 + 6];
      const float x = o0 - (float)h;   // x - gx
      const float y = o1 - (float)w;   // y - gy

      int m = -1;                      // later scatter (i=1) wins
      if (a == anc1 && h == cX1 && w == cY1)      m = 1;
      else if (a == anc0 && h == cX0 && w == cY0) m = 0;

      if (m < 0) {
        // IoU vs both targets -> conf suppression
        const float px1 = x - o2 * 0.5f, px2 = x + o2 * 0.5f;
        const float py1 = y - o3 * 0.5f, py2 = y + o3 * 0.5f;
        const float ap  = o2 * o3;
        float ix = fmaxf(fminf(px2, bw0) - fmaxf(px1, cx0), 0.f);
        float iy = fmaxf(fminf(py2, bh0) - fmaxf(py1, cy0), 0.f);
        float in0 = ix * iy;
        const float iou0 = in0 / (ap + aT0 - in0 + 1e-16f);
        ix = fmaxf(fminf(px2, bw1) - fmaxf(px1, cx1), 0.f);
        iy = fmaxf(fminf(py2, bh1) - fmaxf(py1, cy1), 0.f);
        float in1 = ix * iy;
        const float iou1 = in1 / (ap + aT1 - in1 + 1e-16f);
        if (fmaxf(iou0, iou1) <= 0.5f)        // not suppressed: t=0 BCE term
          s4 -= fmaxf(log1pf(-o4), -100.f);
        // suppressed or not: MSE and cls terms are exactly 0 here
      } else {
        const float tcx = m ? cx1 : cx0, tcy = m ? cy1 : cy0;
        const float tbw = m ? bw1 : bw0, tbh = m ? bh1 : bh0;
        const float tcl = m ? cl1 : cl0;
        const float dx = o0 - tcx;  s0 += dx * dx;   // (x - tX) = o0 - cx
        const float dy = o1 - tcy;  s1 += dy * dy;
        const float dw = o2 - tbw;  s2 += dw * dw;
        const float dh = o3 - tbh;  s3 += dh * dh;
        s4 -= fmaxf(logf(o4), -100.f);               // conf BCE, t=1
        // class BCE: t = tcl for BOTH channels (matched[...,5:6] broadcast)
        s5 -= tcl * fmaxf(logf(o5), -100.f) + (1.f - tcl) * fmaxf(log1pf(-o5), -100.f);
        s5 -= tcl * fmaxf(logf(o6), -100.f) + (1.f - tcl) * fmaxf(log1pf(-o6), -100.f);
      }
    }
  }

  // wave32 reduction via chained WMMA, then cross-wave via LDS
  const float t0 = waveReduce(s0), t1 = waveReduce(s1), t2 = waveReduce(s2);
  const float t3 = waveReduce(s3), t4 = waveReduce(s4), t5 = waveReduce(s5);
  const int lane = tid & 31, wid = tid >> 5;
  if (lane == 0) {
    wsum[wid * 6 + 0] = t0; wsum[wid * 6 + 1] = t1; wsum[wid * 6 + 2] = t2;
    wsum[wid * 6 + 3] = t3; wsum[wid * 6 + 4] = t4; wsum[wid * 6 + 5] = t5;
  }
  __syncthreads();
  if (tid < 6) {
    float acc = 0.f;
#pragma unroll
    for (int k = 0; k < 8; ++k) acc += wsum[k * 6 + tid];
    ws[blockIdx.x * 6 + tid] = acc;    // deterministic store, no atomics
  }
}

// ---------------- kernel 3: fold partials, write 7 outputs ----------------
__global__ __launch_bounds__(256) void yolo_final(const float* __restrict__ ws,
                                                  float* __restrict__ out7) {
  __shared__ float wsum[8 * 6];
  const int tid = threadIdx.x;
  float s[6] = {0.f, 0.f, 0.f, 0.f, 0.f, 0.f};
  for (int i = tid; i < NBLK; i += 256) {      // 6 uniform iterations / thread
#pragma unroll
    for (int j = 0; j < 6; ++j) s[j] += ws[i * 6 + j];
  }
  float r[6];
#pragma unroll
  for (int j = 0; j < 6; ++j) r[j] = waveReduce(s[j]);
  const int lane = tid & 31, wid = tid >> 5;
  if (lane == 0) {
#pragma unroll
    for (int j = 0; j < 6; ++j) wsum[wid * 6 + j] = r[j];
  }
  __syncthreads();
  if (tid == 0) {
    float S[6] = {0.f, 0.f, 0.f, 0.f, 0.f, 0.f};
#pragma unroll
    for (int k = 0; k < 8; ++k)
#pragma unroll
      for (int j = 0; j < 6; ++j) S[j] += wsum[k * 6 + j];
    const float invN1 = 1.f / (float)NCELLS;
    const float invN2 = 1.f / ((float)NCELLS * (float)CC);
    const float xL = S[0] * invN1, yL = S[1] * invN1;
    const float wL = S[2] * invN1, hL = S[3] * invN1;
    const float confL = S[4] * invN1, clsL = S[5] * invN2;
    const float loss = 2.5f * (xL + yL) + 2.5f * (wL + hL) + confL + clsL;
    out7[0] = loss; out7[1] = xL; out7[2] = yL; out7[3] = wL;
    out7[4] = hL;   out7[5] = confL; out7[6] = clsL;
  }
}

// ---------------- launcher ----------------
extern "C" void kernel_launch(void* const* d_in, const int* in_sizes, int n_in,
                              void* d_out, int out_size, void* d_ws, size_t ws_size,
                              hipStream_t stream) {
  const float* output = (const float*)d_in[0];
  const float* boxes  = (const float*)d_in[1];
  const int*   labels = (const int*)d_in[2];
  const float* areas  = (const float*)d_in[3];
  float* ws  = (float*)d_ws;   // [0, NBLK*6): block partials; then 128 target records
  float* out = (float*)d_out;  // 7 floats

  yolo_setup<<<1, 128, 0, stream>>>(boxes, labels, areas, ws);
  yolo_main <<<NBLK, 256, 0, stream>>>(output, ws);
  yolo_final<<<1, 256, 0, stream>>>(ws, out);
}